// Brain_17789754540385
// MI455X (gfx1250) — compile-verified
//
#include <hip/hip_runtime.h>
#include <hip/hip_bf16.h>
#include <stdint.h>

#define UNITS  3
#define GATES  12          // 4 * UNITS, keras gate order i,f,g,o
#define GP     6           // gate pairs (v2f)
#define T_LEN  256
#define TT     16          // timesteps per LDS tile
#define NT     (T_LEN / TT)
#define ROWF   20          // x-row floats in LDS (80B: 16B aligned)
#define HROW   52          // h-row floats in LDS (208B: 16B aligned, 48 used)
#define BLOCK  256

typedef __attribute__((ext_vector_type(2))) float v2f;
typedef __attribute__((ext_vector_type(8))) float v8f;

__device__ __forceinline__ float fast_exp2(float x) { return __builtin_amdgcn_exp2f(x); }
__device__ __forceinline__ float fast_rcp (float x) { return __builtin_amdgcn_rcpf(x); }

#if __has_builtin(__builtin_amdgcn_tanhf)
#define HAVE_TANH 1
__device__ __forceinline__ float tanh_f(float x) { return __builtin_amdgcn_tanhf(x); }
#elif __has_builtin(__builtin_amdgcn_tanh_f32)
#define HAVE_TANH 1
__device__ __forceinline__ float tanh_f(float x) { return __builtin_amdgcn_tanh_f32(x); }
#else
#define HAVE_TANH 0
__device__ __forceinline__ float tanh_f(float x) {
  float e = fast_exp2(2.885390081777927f * x); // e^(2x)
  return 1.0f - 2.0f * fast_rcp(e + 1.0f);
}
#endif

__device__ __forceinline__ float sigm(float x) {
#if HAVE_TANH
  return fmaf(0.5f, tanh_f(0.5f * x), 0.5f);   // 1 TRANS instead of 2
#else
  return fast_rcp(1.0f + fast_exp2(-1.4426950408889634f * x));
#endif
}

// Async-DMA one thread's 16-timestep (64B) row slice into LDS (4 x b128).
__device__ __forceinline__ void async_row16(unsigned lds, const float* g) {
  asm volatile(
      "global_load_async_to_lds_b128 %0, %1, off\n\t"
      "global_load_async_to_lds_b128 %0, %1, off offset:16\n\t"
      "global_load_async_to_lds_b128 %0, %1, off offset:32\n\t"
      "global_load_async_to_lds_b128 %0, %1, off offset:48"
      :: "v"(lds), "v"(g) : "memory");
}

// z accumulated as 6 x v2f -> v_pk_fma_f32 (uniform weights stay SGPR pairs)
template<int DIN>
__device__ __forceinline__ void lstm_step(const float (&xin)[DIN],
                                          const v2f (&W)[DIN][GP],
                                          const v2f (&U)[UNITS][GP],
                                          const v2f (&Bv)[GP],
                                          float (&c)[UNITS], float (&h)[UNITS]) {
  v2f z[GP];
#pragma unroll
  for (int k = 0; k < GP; ++k) z[k] = Bv[k];
#pragma unroll
  for (int j = 0; j < DIN; ++j) {
    v2f xb = { xin[j], xin[j] };
#pragma unroll
    for (int k = 0; k < GP; ++k) z[k] = __builtin_elementwise_fma(xb, W[j][k], z[k]);
  }
#pragma unroll
  for (int j = 0; j < UNITS; ++j) {
    v2f hb = { h[j], h[j] };   // h_{t-1}
#pragma unroll
    for (int k = 0; k < GP; ++k) z[k] = __builtin_elementwise_fma(hb, U[j][k], z[k]);
  }
#pragma unroll
  for (int u = 0; u < UNITS; ++u) {
    float zi = z[(u)     >> 1][(u)     & 1];
    float zf = z[(3 + u) >> 1][(3 + u) & 1];
    float zg = z[(6 + u) >> 1][(6 + u) & 1];
    float zo = z[(9 + u) >> 1][(9 + u) & 1];
    float ig = sigm(zi);
    float fg = sigm(zf);
    float gg = tanh_f(zg);
    float og = sigm(zo);
    float cn = fmaf(fg, c[u], ig * gg);
    c[u] = cn;
    h[u] = og * tanh_f(cn);
  }
}

template<int R>
__device__ __forceinline__ void loadW(v2f (&dst)[R][GP], const float* __restrict__ s) {
#pragma unroll
  for (int j = 0; j < R; ++j)
#pragma unroll
    for (int k = 0; k < GP; ++k)
      dst[j][k] = v2f{ s[j * GATES + 2 * k], s[j * GATES + 2 * k + 1] };
}
__device__ __forceinline__ void loadB(v2f (&dst)[GP], const float* __restrict__ s) {
#pragma unroll
  for (int k = 0; k < GP; ++k) dst[k] = v2f{ s[2 * k], s[2 * k + 1] };
}

// Phase for layer 1: consume x tile from LDS, write h1 tile (48 floats) to hrow.
__device__ __forceinline__ void phase_first(const v2f (&W)[1][GP], const v2f (&U)[UNITS][GP],
                                            const v2f (&Bv)[GP], float (&c)[UNITS], float (&h)[UNITS],
                                            const float* xrow, float* hrow) {
#pragma unroll
  for (int g = 0; g < 4; ++g) {
    float4 xg = *(const float4*)(xrow + 4 * g);
    float hh[12];
    float xv[1];
    xv[0] = xg.x; lstm_step<1>(xv, W, U, Bv, c, h); hh[0] = h[0]; hh[1]  = h[1]; hh[2]  = h[2];
    xv[0] = xg.y; lstm_step<1>(xv, W, U, Bv, c, h); hh[3] = h[0]; hh[4]  = h[1]; hh[5]  = h[2];
    xv[0] = xg.z; lstm_step<1>(xv, W, U, Bv, c, h); hh[6] = h[0]; hh[7]  = h[1]; hh[8]  = h[2];
    xv[0] = xg.w; lstm_step<1>(xv, W, U, Bv, c, h); hh[9] = h[0]; hh[10] = h[1]; hh[11] = h[2];
    ((float4*)(hrow + 12 * g))[0] = make_float4(hh[0], hh[1], hh[2],  hh[3]);
    ((float4*)(hrow + 12 * g))[1] = make_float4(hh[4], hh[5], hh[6],  hh[7]);
    ((float4*)(hrow + 12 * g))[2] = make_float4(hh[8], hh[9], hh[10], hh[11]);
  }
}

// Phase for layers 2..3: read h tile, step, overwrite in place.
__device__ __forceinline__ void phase_mid(const v2f (&W)[UNITS][GP], const v2f (&U)[UNITS][GP],
                                          const v2f (&Bv)[GP], float (&c)[UNITS], float (&h)[UNITS],
                                          float* hrow) {
#pragma unroll
  for (int g = 0; g < 4; ++g) {
    float4 a = ((const float4*)(hrow + 12 * g))[0];
    float4 b = ((const float4*)(hrow + 12 * g))[1];
    float4 d = ((const float4*)(hrow + 12 * g))[2];
    float hh[12] = { a.x,a.y,a.z,a.w, b.x,b.y,b.z,b.w, d.x,d.y,d.z,d.w };
#pragma unroll
    for (int s = 0; s < 4; ++s) {
      float xin[3] = { hh[3*s], hh[3*s+1], hh[3*s+2] };
      lstm_step<3>(xin, W, U, Bv, c, h);
      hh[3*s] = h[0]; hh[3*s+1] = h[1]; hh[3*s+2] = h[2];
    }
    ((float4*)(hrow + 12 * g))[0] = make_float4(hh[0], hh[1], hh[2],  hh[3]);
    ((float4*)(hrow + 12 * g))[1] = make_float4(hh[4], hh[5], hh[6],  hh[7]);
    ((float4*)(hrow + 12 * g))[2] = make_float4(hh[8], hh[9], hh[10], hh[11]);
  }
}

// Phase for layer 4: read h tile, step, no write-back (only carry matters).
__device__ __forceinline__ void phase_last(const v2f (&W)[UNITS][GP], const v2f (&U)[UNITS][GP],
                                           const v2f (&Bv)[GP], float (&c)[UNITS], float (&h)[UNITS],
                                           const float* hrow) {
#pragma unroll
  for (int g = 0; g < 4; ++g) {
    float4 a = ((const float4*)(hrow + 12 * g))[0];
    float4 b = ((const float4*)(hrow + 12 * g))[1];
    float4 d = ((const float4*)(hrow + 12 * g))[2];
    float hh[12] = { a.x,a.y,a.z,a.w, b.x,b.y,b.z,b.w, d.x,d.y,d.z,d.w };
#pragma unroll
    for (int s = 0; s < 4; ++s) {
      float xin[3] = { hh[3*s], hh[3*s+1], hh[3*s+2] };
      lstm_step<3>(xin, W, U, Bv, c, h);
    }
  }
}

__global__ __launch_bounds__(BLOCK, 1)
void lstm4_dense_kernel(const float* __restrict__ state,
                        const float* __restrict__ w1, const float* __restrict__ u1, const float* __restrict__ b1,
                        const float* __restrict__ w2, const float* __restrict__ u2, const float* __restrict__ b2,
                        const float* __restrict__ w3, const float* __restrict__ u3, const float* __restrict__ b3,
                        const float* __restrict__ w4, const float* __restrict__ u4, const float* __restrict__ b4,
                        const float* __restrict__ wd, const float* __restrict__ bd,
                        float* __restrict__ out) {
  __shared__ __align__(16) float xs[2][BLOCK * ROWF];
  __shared__ __align__(16) float hb[BLOCK][HROW];

  const int tid = threadIdx.x;
  const int row = blockIdx.x * BLOCK + tid;
  const float* gx = state + (size_t)row * T_LEN;

  unsigned lbuf0 = (unsigned)(uintptr_t)(&xs[0][tid * ROWF]);
  unsigned lbuf1 = (unsigned)(uintptr_t)(&xs[1][tid * ROWF]);

  // ---- uniform weights (SGPR-resident; per-phase clustered use) ----
  v2f W1[1][GP], U1[UNITS][GP], B1[GP];
  v2f W2[UNITS][GP], U2[UNITS][GP], B2[GP];
  v2f W3[UNITS][GP], U3[UNITS][GP], B3[GP];
  v2f W4[UNITS][GP], U4[UNITS][GP], B4[GP];
  loadW<1>(W1, w1);      loadW<UNITS>(U1, u1); loadB(B1, b1);
  loadW<UNITS>(W2, w2);  loadW<UNITS>(U2, u2); loadB(B2, b2);
  loadW<UNITS>(W3, w3);  loadW<UNITS>(U3, u3); loadB(B3, b3);
  loadW<UNITS>(W4, w4);  loadW<UNITS>(U4, u4); loadB(B4, b4);
  const float wd0 = wd[0], wd1 = wd[1], wd2 = wd[2], bd0 = bd[0];

  float c1[UNITS] = {0,0,0}, h1[UNITS] = {0,0,0};
  float c2[UNITS] = {0,0,0}, h2[UNITS] = {0,0,0};
  float c3[UNITS] = {0,0,0}, h3[UNITS] = {0,0,0};
  float c4[UNITS] = {0,0,0}, h4[UNITS] = {0,0,0};

  // prime the x double buffer
  async_row16(lbuf0, gx);

#pragma unroll 1
  for (int tile = 0; tile < NT; ++tile) {
    if (tile + 1 < NT) {
      async_row16((tile & 1) ? lbuf0 : lbuf1, gx + (tile + 1) * TT);
      asm volatile("s_wait_asynccnt 0x4" ::: "memory");  // current tile's 4 done
    } else {
      asm volatile("s_wait_asynccnt 0x0" ::: "memory");
    }
    const float* xrow = &xs[tile & 1][tid * ROWF];
    float* hrow = &hb[tid][0];
    // Layer-phased over the 16-step tile: clusters each layer's uniform
    // weights into one region (SGPR-friendly), carries (h,c) in registers.
    // Each thread touches only its own LDS row -> no barriers.
    phase_first(W1, U1, B1, c1, h1, xrow, hrow);
    phase_mid  (W2, U2, B2, c2, h2, hrow);
    phase_mid  (W3, U3, B3, c3, h3, hrow);
    phase_last (W4, U4, B4, c4, h4, hrow);
  }

  // ---- dense head: out[row] = h4 . wd + bd via 2x v_wmma_f32_16x16x4_f32 ----
  const int  lane = tid & 31;
  const bool hi   = lane >= 16;
  const int  l16  = lane & 15;

  v2f bmat; bmat.x = hi ? wd2 : wd0; bmat.y = hi ? 0.0f : wd1;   // K={0,1}|{2,3}
  v8f cmat;
#pragma unroll
  for (int k = 0; k < 8; ++k) cmat[k] = bd0;

  // WMMA #0: rows = lanes 0..15 of this wave
  float s2 = __shfl(h4[2], l16, 32);
  v2f a0; a0.x = hi ? s2 : h4[0]; a0.y = hi ? 0.0f : h4[1];
  v8f d0 = __builtin_amdgcn_wmma_f32_16x16x4_f32(false, a0, false, bmat,
                                                 (short)0, cmat, false, false);
  // WMMA #1: rows = lanes 16..31 of this wave
  float s0 = __shfl(h4[0], l16 + 16, 32);
  float s1 = __shfl(h4[1], l16 + 16, 32);
  v2f a1; a1.x = hi ? h4[2] : s0; a1.y = hi ? 0.0f : s1;
  v8f d1 = __builtin_amdgcn_wmma_f32_16x16x4_f32(false, a1, false, bmat,
                                                 (short)0, cmat, false, false);

  // D layout: VGPR j -> row j (lanes 0-15) / row j+8 (lanes 16-31), all columns
  // identical. Lanes 0 and 16 write 8 rows each.
  if (l16 == 0) {
    float* o = out + (row - lane) + (hi ? 8 : 0);
#pragma unroll
    for (int j = 0; j < 8; ++j) o[j] = d0[j];
    o += 16;
#pragma unroll
    for (int j = 0; j < 8; ++j) o[j] = d1[j];
  }
}

extern "C" void kernel_launch(void* const* d_in, const int* in_sizes, int n_in,
                              void* d_out, int out_size, void* d_ws, size_t ws_size,
                              hipStream_t stream) {
  const float* state = (const float*)d_in[0];
  const float* w1 = (const float*)d_in[1];
  const float* u1 = (const float*)d_in[2];
  const float* b1 = (const float*)d_in[3];
  const float* w2 = (const float*)d_in[4];
  const float* u2 = (const float*)d_in[5];
  const float* b2 = (const float*)d_in[6];
  const float* w3 = (const float*)d_in[7];
  const float* u3 = (const float*)d_in[8];
  const float* b3 = (const float*)d_in[9];
  const float* w4 = (const float*)d_in[10];
  const float* u4 = (const float*)d_in[11];
  const float* b4 = (const float*)d_in[12];
  const float* wd = (const float*)d_in[13];
  const float* bd = (const float*)d_in[14];
  float* out = (float*)d_out;

  const int Bn = in_sizes[0] / T_LEN;   // 8192
  dim3 grid(Bn / BLOCK), block(BLOCK);
  lstm4_dense_kernel<<<grid, block, 0, stream>>>(state,
                                                 w1, u1, b1, w2, u2, b2,
                                                 w3, u3, b3, w4, u4, b4,
                                                 wd, bd, out);
}